// INSTA_3350074491542
// MI455X (gfx1250) — compile-verified
//
#include <hip/hip_runtime.h>
#include <cmath>

// ---------------- problem constants ----------------
#define CC   512
#define SSZ  11
#define PP   121     // S*S
#define KKK  9       // K*K
#define BB   128
#define O1   1024    // 2*C
#define HID  32      // C/SIGMA
#define INV_SQ 0.99999500003749981f   // 1/sqrt(1+1e-5)

// ---------------- WMMA f32 16x16x4 ----------------
typedef __attribute__((ext_vector_type(2))) float v2f;
typedef __attribute__((ext_vector_type(8))) float v8f;

__device__ __forceinline__ v8f wmma_f32_16x16x4(v2f a, v2f b, v8f c) {
#if __has_builtin(__builtin_amdgcn_wmma_f32_16x16x4_f32)
  return __builtin_amdgcn_wmma_f32_16x16x4_f32(false, a, false, b, (short)0, c,
                                               false, false);
#else
  c[0] += a[0] * b[0];   // placeholder; histogram shows wmma==0 if hit
  return c;
#endif
}

// ---------------- Tensor Data Mover (gfx1250 TDM) ----------------
#if __has_builtin(__builtin_amdgcn_tensor_load_to_lds) && \
    __has_builtin(__builtin_amdgcn_s_wait_tensorcnt)
#define CDNA5_HAS_TDM 1
typedef __attribute__((ext_vector_type(4))) unsigned int u32x4;
typedef __attribute__((ext_vector_type(8))) int i32x8;
typedef __attribute__((ext_vector_type(4))) int i32x4;

// D# group1 dword0 flags: data_size=2 (4B elements) [+ optional LDS padding]
#define TDM_F32            0x00020000u
#define TDM_F32_PAD1_EV32  (0x00020000u | (1u << 20) | (4u << 22)) // +1 dw per 32 dw

// 2D tile DMA: global (tile_d1 rows x tile_d0 elems, row stride stride0) -> LDS.
// OOB columns (x >= tensor_d0) read as zero.  Packing per CDNA5 ISA.
__device__ __forceinline__ void tdm_load_2d(unsigned lds_off, const void* gptr,
                                            unsigned tensor_d0, unsigned tensor_d1,
                                            unsigned tile_d0, unsigned tile_d1,
                                            unsigned long long stride0,
                                            unsigned flags) {
  unsigned long long ga = (unsigned long long)(size_t)gptr;
  u32x4 g0;
  g0.x = 1u;                                            // count=1, user mode
  g0.y = lds_off;                                       // lds_addr (bytes)
  g0.z = (unsigned)(ga & 0xFFFFFFFFu);                  // global_addr[31:0]
  g0.w = (unsigned)((ga >> 32) & 0x01FFFFFFu) | (2u << 30); // addr[56:32] | type=2
  i32x8 g1;
  g1[0] = (int)flags;                                   // wg_mask=0 | data_size | pad cfg
  g1[1] = (int)((tensor_d0 & 0xFFFFu) << 16);           // abar=0 | tensor_dim0[15:0]
  g1[2] = (int)(((tensor_d0 >> 16) & 0xFFFFu) | ((tensor_d1 & 0xFFFFu) << 16));
  g1[3] = (int)(((tensor_d1 >> 16) & 0xFFFFu) | ((tile_d0 & 0xFFFFu) << 16));
  g1[4] = (int)(tile_d1 & 0xFFFFu);                     // tile_dim1 | tile_dim2=0
  g1[5] = (int)(unsigned)(stride0 & 0xFFFFFFFFu);       // tensor_dim0_stride[31:0]
  g1[6] = (int)(unsigned)((stride0 >> 32) & 0xFFFFu);   // stride0[47:32] | stride1=0
  g1[7] = 0;
  i32x4 z4 = {0, 0, 0, 0};                              // groups 2/3 unused (2D)
  i32x8 z8 = {0, 0, 0, 0, 0, 0, 0, 0};
  __builtin_amdgcn_tensor_load_to_lds(g0, g1, z4, z4, z8, 0);  // 6-arg clang-23 form
}
#endif

// ---------------- small kernels ----------------

__global__ void k_dct_pool(const float* __restrict__ f, const float* __restrict__ dct,
                           float* __restrict__ y, int batch) {
  int wid = (blockIdx.x * blockDim.x + threadIdx.x) >> 5;
  int lane = threadIdx.x & 31;
  if (wid >= batch * CC) return;
  int c = wid % CC;
  const float* fp = f + (size_t)wid * PP;
  const float* dp = dct + (size_t)c * PP;
  float acc = 0.f;
  for (int p = lane; p < PP; p += 32) acc += fp[p] * dp[p];
  for (int off = 16; off; off >>= 1) acc += __shfl_xor(acc, off, 32);
  if (lane == 0) y[wid] = acc;
}

__global__ void k_mlp_ck(const float* __restrict__ y, const float* __restrict__ fc1,
                         const float* __restrict__ fc2, const float* __restrict__ g_ch,
                         const float* __restrict__ b_ch, float* __restrict__ ck) {
  __shared__ float hid[HID];
  int b = blockIdx.x;
  int t = threadIdx.x;
  const float* yb = y + (size_t)b * CC;
  if (t < HID) {
    float acc = 0.f;
    const float* w = fc1 + (size_t)t * CC;
    for (int k = 0; k < CC; ++k) acc += w[k] * yb[k];
    hid[t] = fmaxf(acc, 0.f);
  }
  __syncthreads();
  for (int i = t; i < CC * KKK; i += blockDim.x) {
    const float* w = fc2 + (size_t)i * HID;
    float acc = 0.f;
#pragma unroll
    for (int j = 0; j < HID; ++j) acc += w[j] * hid[j];
    float sig = 1.f / (1.f + expf(-acc));
    int c = i / KKK;
    ck[(size_t)b * CC * KKK + i] = sig * (g_ch[c] * INV_SQ) + b_ch[c];
  }
}

__global__ void k_spatial(const float* __restrict__ f, const float* __restrict__ w_conv,
                          const float* __restrict__ b_conv, const float* __restrict__ g_sp,
                          const float* __restrict__ b_sp, float* __restrict__ sk, int batch) {
  int wid = (blockIdx.x * blockDim.x + threadIdx.x) >> 5;
  int lane = threadIdx.x & 31;
  if (wid >= batch * PP) return;
  int b = wid / PP, p = wid % PP;
  float acc[KKK];
#pragma unroll
  for (int kk = 0; kk < KKK; ++kk) acc[kk] = 0.f;
  for (int c = lane; c < CC; c += 32) {
    float xv = f[((size_t)b * CC + c) * PP + p];
#pragma unroll
    for (int kk = 0; kk < KKK; ++kk) acc[kk] += xv * w_conv[kk * CC + c];
  }
#pragma unroll
  for (int kk = 0; kk < KKK; ++kk)
    for (int off = 16; off; off >>= 1) acc[kk] += __shfl_xor(acc[kk], off, 32);
  if (lane == 0) {
    float s = g_sp[p] * INV_SQ, sh = b_sp[p];
#pragma unroll
    for (int kk = 0; kk < KKK; ++kk)
      sk[(size_t)wid * KKK + kk] = (acc[kk] + b_conv[kk]) * s + sh;
  }
}

__global__ void k_zero(float* __restrict__ p, int n) {
  int i = blockIdx.x * blockDim.x + threadIdx.x;
  if (i < n) p[i] = 0.f;
}

// ---------------- WMMA GEMM: Y = act(bn(W @ X + bias)) ----------------
// W:(M,Kdim) row-major. X:(batch,Kdim,PP). accumulate=1 -> atomicAdd into (M,PP).
// A tiles staged in LDS as [m][33] (padded), B tiles as [k][64].
#define TM 128
#define TN 64
#define TK 32
#define LDA 33

__global__ __launch_bounds__(256)
void k_gemm_bn_act(const float* __restrict__ W, const float* __restrict__ X,
                   const float* __restrict__ bias, const float* __restrict__ gamma,
                   const float* __restrict__ beta, float* __restrict__ Y,
                   int M, int Kdim, int act, int accumulate) {
  __shared__ float As[2][TM * LDA];
  __shared__ float Bs[2][TK * TN];

  const int t    = threadIdx.x;
  const int lane = t & 31;
  const int m0   = blockIdx.x * TM;
  const int p0   = blockIdx.y * TN;
  const int bz   = blockIdx.z;
  const float* Xb = X + (size_t)bz * Kdim * PP;

#ifdef CDNA5_HAS_TDM
  const int wave = __builtin_amdgcn_readfirstlane(t >> 5);
#else
  const int wave = t >> 5;
#endif
  const int wm  = (wave & 3) * 32;   // wave m-offset in block tile
  const int wn  = (wave >> 2) * 32;  // wave n-offset in block tile
  const int hl  = lane >> 4;         // K-pair select (lanes 16-31 hold K+2,K+3)
  const int l16 = lane & 15;

  v8f c00 = {}, c01 = {}, c10 = {}, c11 = {};

  // ---- tile staging: TDM DMA (async, per-wave slices) or manual fallback ----
#ifdef CDNA5_HAS_TDM
  #define STAGE_TILES(KB, BUF)                                                   \
    do {                                                                         \
      /* A slice: 16 rows x 32 k, pad 1 dword per row -> LDS stride 33 */        \
      tdm_load_2d((unsigned)(size_t)(&As[BUF][wave * 16 * LDA]),                 \
                  W + (size_t)(m0 + wave * 16) * Kdim + (KB),                    \
                  32u, 16u, 32u, 16u, (unsigned long long)Kdim,                  \
                  TDM_F32_PAD1_EV32);                                            \
      /* B slice: 4 k-rows x 64 p, zero-fill p >= PP-p0 */                       \
      tdm_load_2d((unsigned)(size_t)(&Bs[BUF][wave * 4 * TN]),                   \
                  Xb + (size_t)((KB) + wave * 4) * PP + p0,                      \
                  (unsigned)(PP - p0), 4u, (unsigned)TN, 4u,                     \
                  (unsigned long long)PP, TDM_F32);                              \
    } while (0)
  #define WAIT_TILES() __builtin_amdgcn_s_wait_tensorcnt(0)
#else
  #define STAGE_TILES(KB, BUF)                                                   \
    do {                                                                         \
      int row  = t >> 1;                                                         \
      int colb = (t & 1) * 16;                                                   \
      const float* srcA = W + (size_t)(m0 + row) * Kdim + (KB) + colb;           \
      float* dstA = &As[BUF][row * LDA + colb];                                  \
      _Pragma("unroll")                                                          \
      for (int q = 0; q < 4; ++q) {                                              \
        float4 v = *(const float4*)(srcA + q * 4);                               \
        dstA[q * 4 + 0] = v.x; dstA[q * 4 + 1] = v.y;                            \
        dstA[q * 4 + 2] = v.z; dstA[q * 4 + 3] = v.w;                            \
      }                                                                          \
      int kr = t >> 3;                                                           \
      int pb = (t & 7) * 8;                                                      \
      const float* srcB = Xb + (size_t)((KB) + kr) * PP + p0;                    \
      _Pragma("unroll")                                                          \
      for (int q = 0; q < 8; ++q) {                                              \
        int p = pb + q;                                                          \
        Bs[BUF][kr * TN + p] = (p0 + p < PP) ? srcB[p] : 0.f;                    \
      }                                                                          \
    } while (0)
  #define WAIT_TILES() do {} while (0)
#endif

  STAGE_TILES(0, 0);
  WAIT_TILES();
  __syncthreads();

  for (int kb = 0; kb < Kdim; kb += TK) {
    const int cur = (kb / TK) & 1;
    const bool more = (kb + TK) < Kdim;
    if (more) STAGE_TILES(kb + TK, cur ^ 1);   // DMA next chunk, overlap compute

#pragma unroll
    for (int kk = 0; kk < TK; kk += 4) {
      int kc = kk + hl * 2;
      const float* ar0 = &As[cur][(wm + l16) * LDA + kc];
      const float* ar1 = &As[cur][(wm + 16 + l16) * LDA + kc];
      const float* bc0 = &Bs[cur][kc * TN + wn + l16];
      const float* bc1 = bc0 + 16;
      v2f a0; a0[0] = ar0[0]; a0[1] = ar0[1];
      v2f a1; a1[0] = ar1[0]; a1[1] = ar1[1];
      v2f b0; b0[0] = bc0[0]; b0[1] = bc0[TN];
      v2f b1; b1[0] = bc1[0]; b1[1] = bc1[TN];
      c00 = wmma_f32_16x16x4(a0, b0, c00);
      c01 = wmma_f32_16x16x4(a0, b1, c01);
      c10 = wmma_f32_16x16x4(a1, b0, c10);
      c11 = wmma_f32_16x16x4(a1, b1, c11);
    }

    if (more) WAIT_TILES();
    __syncthreads();
  }

  // ---- epilogue: bias + BN + activation (+ optional batch-sum accumulate) ----
#pragma unroll
  for (int i = 0; i < 2; ++i) {
#pragma unroll
    for (int j = 0; j < 2; ++j) {
      v8f cv = (i == 0) ? (j == 0 ? c00 : c01) : (j == 0 ? c10 : c11);
      int p = p0 + wn + j * 16 + l16;
      if (p >= PP) continue;
#pragma unroll
      for (int r = 0; r < 8; ++r) {
        int m = m0 + wm + i * 16 + r + hl * 8;
        float v = cv[r];
        v = (v + bias[m]) * (gamma[m] * INV_SQ) + beta[m];
        if (act == 1)      v = fmaxf(v, 0.f);
        else if (act == 2) v = 1.f / (1.f + expf(-v));
        if (accumulate) atomicAdd(&Y[(size_t)m * PP + p], v);
        else            Y[((size_t)bz * M + m) * PP + p] = v;
      }
    }
  }
}

// task_kernel[c,p,kk] = sk_t[p,kk] * ck_t[c,kk]  (written into d_out slice)
__global__ void k_taskkernel(const float* __restrict__ sk_t, const float* __restrict__ ck_t,
                             float* __restrict__ tk) {
  int i = blockIdx.x * blockDim.x + threadIdx.x;
  if (i >= CC * PP * KKK) return;
  int kk = i % KKK;
  int p  = (i / KKK) % PP;
  int c  = i / (KKK * PP);
  tk[i] = sk_t[p * KKK + kk] * ck_t[c * KKK + kk];
}

// out = x + mean_kk( unfold(x) * task_kernel * sk * ck )
__global__ void k_adapt(const float* __restrict__ x, const float* __restrict__ tk,
                        const float* __restrict__ sk, const float* __restrict__ ck,
                        float* __restrict__ out) {
  int i = blockIdx.x * blockDim.x + threadIdx.x;
  if (i >= BB * CC * PP) return;
  int p = i % PP;
  int c = (i / PP) % CC;
  int b = i / (PP * CC);
  int h = p / SSZ, w = p % SSZ;
  const float* xb  = x  + ((size_t)b * CC + c) * PP;
  const float* tkp = tk + ((size_t)c * PP + p) * KKK;
  const float* skp = sk + ((size_t)b * PP + p) * KKK;
  const float* ckp = ck + ((size_t)b * CC + c) * KKK;
  float acc = 0.f;
#pragma unroll
  for (int ki = 0; ki < 3; ++ki) {
    int hh = h + ki - 1;
#pragma unroll
    for (int kj = 0; kj < 3; ++kj) {
      int ww = w + kj - 1;
      int kk = ki * 3 + kj;
      float xv = (hh >= 0 && hh < SSZ && ww >= 0 && ww < SSZ) ? xb[hh * SSZ + ww] : 0.f;
      acc += xv * tkp[kk] * skp[kk] * ckp[kk];
    }
  }
  out[i] = x[i] + acc * (1.f / 9.f);
}

// ---------------- driver ----------------
extern "C" void kernel_launch(void* const* d_in, const int* in_sizes, int n_in,
                              void* d_out, int out_size, void* d_ws, size_t ws_size,
                              hipStream_t stream) {
  (void)in_sizes; (void)n_in; (void)out_size; (void)ws_size;
  const float* x      = (const float*)d_in[0];
  const float* dct_w  = (const float*)d_in[1];
  const float* w_conv = (const float*)d_in[2];
  const float* b_conv = (const float*)d_in[3];
  const float* g_sp   = (const float*)d_in[4];
  const float* b_sp   = (const float*)d_in[5];
  const float* g_ch   = (const float*)d_in[6];
  const float* b_ch   = (const float*)d_in[7];
  const float* fc1    = (const float*)d_in[8];
  const float* fc2    = (const float*)d_in[9];
  const float* up_w1  = (const float*)d_in[10];
  const float* up_b1  = (const float*)d_in[11];
  const float* up_g1  = (const float*)d_in[12];
  const float* up_bb1 = (const float*)d_in[13];
  const float* up_w2  = (const float*)d_in[14];
  const float* up_b2  = (const float*)d_in[15];
  const float* up_g2  = (const float*)d_in[16];
  const float* up_bb2 = (const float*)d_in[17];
  const float* lo_w1  = (const float*)d_in[18];
  const float* lo_b1  = (const float*)d_in[19];
  const float* lo_g1  = (const float*)d_in[20];
  const float* lo_bb1 = (const float*)d_in[21];
  const float* lo_w2  = (const float*)d_in[22];
  const float* lo_b2  = (const float*)d_in[23];
  const float* lo_g2  = (const float*)d_in[24];
  const float* lo_bb2 = (const float*)d_in[25];

  float* out    = (float*)d_out;
  float* out_tk = out + (size_t)BB * CC * PP;   // task_kernel slice of d_out

  float* ws = (float*)d_ws;
  size_t off = 0;
  auto carve = [&](size_t n) { float* p = ws + off; off += (n + 63) & ~(size_t)63; return p; };
  float* y    = carve((size_t)BB * CC);
  float* ck   = carve((size_t)BB * CC * KKK);
  float* sk   = carve((size_t)BB * PP * KKK);
  float* a1   = carve((size_t)BB * O1 * PP);
  float* asum = carve((size_t)O1 * PP);
  float* a3   = carve((size_t)O1 * PP);
  float* tsk  = carve((size_t)CC * PP);
  float* y_t  = carve((size_t)CC);
  float* ck_t = carve((size_t)CC * KKK);
  float* sk_t = carve((size_t)PP * KKK);

  // instance branch (attention kernels)
  k_dct_pool<<<(BB * CC) / 8, 256, 0, stream>>>(x, dct_w, y, BB);
  k_mlp_ck<<<BB, 256, 0, stream>>>(y, fc1, fc2, g_ch, b_ch, ck);
  k_spatial<<<(BB * PP + 7) / 8, 256, 0, stream>>>(x, w_conv, b_conv, g_sp, b_sp, sk, BB);

  // CLM task branch: big WMMA GEMMs (TDM-staged, double-buffered)
  k_gemm_bn_act<<<dim3(O1 / TM, 2, BB), 256, 0, stream>>>(up_w1, x, up_b1, up_g1, up_bb1,
                                                          a1, O1, CC, 1, 0);
  k_zero<<<(O1 * PP + 255) / 256, 256, 0, stream>>>(asum, O1 * PP);
  k_gemm_bn_act<<<dim3(O1 / TM, 2, BB), 256, 0, stream>>>(up_w2, a1, up_b2, up_g2, up_bb2,
                                                          asum, O1, O1, 1, 1);   // fused batch-sum
  k_gemm_bn_act<<<dim3(O1 / TM, 2, 1), 256, 0, stream>>>(lo_w1, asum, lo_b1, lo_g1, lo_bb1,
                                                         a3, O1, O1, 1, 0);
  k_gemm_bn_act<<<dim3(CC / TM, 2, 1), 256, 0, stream>>>(lo_w2, a3, lo_b2, lo_g2, lo_bb2,
                                                         tsk, CC, O1, 2, 0);     // sigmoid

  // task kernels from task_s
  k_dct_pool<<<CC / 8, 256, 0, stream>>>(tsk, dct_w, y_t, 1);
  k_mlp_ck<<<1, 256, 0, stream>>>(y_t, fc1, fc2, g_ch, b_ch, ck_t);
  k_spatial<<<(PP + 7) / 8, 256, 0, stream>>>(tsk, w_conv, b_conv, g_sp, b_sp, sk_t, 1);
  k_taskkernel<<<(CC * PP * KKK + 255) / 256, 256, 0, stream>>>(sk_t, ck_t, out_tk);

  // final fused unfold * (task_kernel * inst) mean + residual
  k_adapt<<<(BB * CC * PP + 255) / 256, 256, 0, stream>>>(x, out_tk, sk, ck, out);
}